// GCNModel_2774548873761
// MI455X (gfx1250) — compile-verified
//
#include <hip/hip_runtime.h>
#include <hip/hip_bf16.h>

// ---------------------------------------------------------------------------
// GCN (2x GCNConv + FC) for MI455X / gfx1250, wave32, WMMA f32 16x16x4.
//
// Pipeline (all on `stream`, in order):
//   1) deg[i] = 1 (self-loop) ; deg[col[e]] += w[e]     (atomics)
//   2) dinv[i] = rsqrt(deg[i])
//   3) h  = x @ W1                                      (WMMA f32)
//   4) agg = 0 ; agg[col] += norm * h[row]              (edge scatter, atomics)
//   5) agg = relu(agg + dinv^2 * h + b1)                (self-loop + bias)
//   6) h  = agg @ W2                                    (WMMA f32)
//   7) agg = 0 ; scatter ; agg = relu(agg + dinv^2*h + b2)
//   8) out = agg @ Wfc + bfc                            (tiny FC)
// ---------------------------------------------------------------------------

#define N_NODES 100000
#define N_EDGES 1600000
#define IN_DIM  512
#define HID     64
#define OUT_DIM 5

typedef __attribute__((ext_vector_type(2))) float v2f;
typedef __attribute__((ext_vector_type(8))) float v8f;

// ---------------------------------------------------------------------------
// Degree init: deg[i] = 1.0f (self-loop weight)
__global__ void k_init_deg(float* __restrict__ deg) {
    int i = blockIdx.x * blockDim.x + threadIdx.x;
    if (i < N_NODES) deg[i] = 1.0f;
}

// deg[col[e]] += w[e]
__global__ void k_deg_scatter(const int* __restrict__ col,
                              const float* __restrict__ w,
                              float* __restrict__ deg) {
    int stride = gridDim.x * blockDim.x;
    for (int e = blockIdx.x * blockDim.x + threadIdx.x; e < N_EDGES; e += stride)
        atomicAdd(deg + col[e], w[e]);
}

// dinv[i] = deg > 0 ? rsqrt(deg) : 0    (in place)
__global__ void k_dinv(float* __restrict__ deg) {
    int i = blockIdx.x * blockDim.x + threadIdx.x;
    if (i < N_NODES) {
        float d = deg[i];
        deg[i] = (d > 0.0f) ? rsqrtf(d) : 0.0f;
    }
}

// Zero a float buffer of n elements.
__global__ void k_zero(float* __restrict__ p, int n) {
    int stride = gridDim.x * blockDim.x;
    for (int i = blockIdx.x * blockDim.x + threadIdx.x; i < n; i += stride)
        p[i] = 0.0f;
}

// ---------------------------------------------------------------------------
// GEMM: H[M x 64] = X[M x K] * W[K x 64], fp32 via V_WMMA_F32_16X16X4_F32.
// One wave -> 16 rows x 64 cols (4 accumulators of v8f).
// M (=N_NODES) is a multiple of 16; K is a multiple of 4.
//
// A-frag (16x4 f32, 2 VGPRs / lane):
//   lanes 0-15  : row M = lane,     K = {k, k+1}
//   lanes 16-31 : row M = lane-16,  K = {k+2, k+3}
// B-frag (4x16 f32, 2 VGPRs / lane), rows striped across lanes:
//   lanes 0-15  : N = lane,     K = {k,   k+1}
//   lanes 16-31 : N = lane-16,  K = {k+2, k+3}
// C/D (16x16 f32, 8 VGPRs): vgpr j -> row m0 + j + (lane>=16 ? 8 : 0), col lane&15.
__global__ void k_gemm_n64(const float* __restrict__ X,
                           const float* __restrict__ W,
                           float* __restrict__ H,
                           int M, int K) {
    const int lane  = threadIdx.x & 31;
    const int waveT = (blockIdx.x * blockDim.x + threadIdx.x) >> 5;  // wave tile id
    const int nTiles = M >> 4;
    if (waveT >= nTiles) return;

    const int m0   = waveT << 4;
    const int mRow = m0 + (lane & 15);
    const int kSub = (lane & 16) ? 2 : 0;   // K sub-offset for this half-wave
    const int nCol = lane & 15;

    v8f c0 = {}, c1 = {}, c2 = {}, c3 = {};

    for (int k = 0; k < K; k += 4) {
        // A fragment: two consecutive K values for this lane's row.
        v2f a = *(const v2f*)(X + (size_t)mRow * K + (k + kSub));

        // B fragments for the 4 N-tiles.
        const float* wr0 = W + (size_t)(k + kSub)     * HID + nCol;
        const float* wr1 = W + (size_t)(k + kSub + 1) * HID + nCol;
        v2f b0, b1, b2, b3;
        b0.x = wr0[0];  b0.y = wr1[0];
        b1.x = wr0[16]; b1.y = wr1[16];
        b2.x = wr0[32]; b2.y = wr1[32];
        b3.x = wr0[48]; b3.y = wr1[48];

        c0 = __builtin_amdgcn_wmma_f32_16x16x4_f32(false, a, false, b0,
                                                   (short)0, c0, false, false);
        c1 = __builtin_amdgcn_wmma_f32_16x16x4_f32(false, a, false, b1,
                                                   (short)0, c1, false, false);
        c2 = __builtin_amdgcn_wmma_f32_16x16x4_f32(false, a, false, b2,
                                                   (short)0, c2, false, false);
        c3 = __builtin_amdgcn_wmma_f32_16x16x4_f32(false, a, false, b3,
                                                   (short)0, c3, false, false);
    }

    // Store: vgpr j holds row m0 + j (+8 for upper half-wave), col nCol (+16*tile).
    const int rBase = m0 + ((lane & 16) ? 8 : 0);
    #pragma unroll
    for (int j = 0; j < 8; ++j) {
        float* dst = H + (size_t)(rBase + j) * HID + nCol;
        dst[0]  = c0[j];
        dst[16] = c1[j];
        dst[32] = c2[j];
        dst[48] = c3[j];
    }
}

// ---------------------------------------------------------------------------
// Edge scatter: agg[col] += dinv[row] * w * dinv[col] * h[row][:]
// One wave per edge iteration: 32 lanes x float2 = 64 features.
__global__ void k_edge_scatter(const int* __restrict__ row,
                               const int* __restrict__ col,
                               const float* __restrict__ w,
                               const float* __restrict__ dinv,
                               const float* __restrict__ h,
                               float* __restrict__ agg) {
    const int lane   = threadIdx.x & 31;
    const int wid    = (blockIdx.x * blockDim.x + threadIdx.x) >> 5;
    const int nWaves = (gridDim.x * blockDim.x) >> 5;

    for (int e = wid; e < N_EDGES; e += nWaves) {
        const int r = row[e];
        const int c = col[e];
        const float norm = dinv[r] * w[e] * dinv[c];
        v2f hv = *(const v2f*)(h + (size_t)r * HID + lane * 2);
        float* dst = agg + (size_t)c * HID + lane * 2;
        atomicAdd(dst + 0, norm * hv.x);
        atomicAdd(dst + 1, norm * hv.y);
    }
}

// ---------------------------------------------------------------------------
// Finalize: agg = act(agg + dinv^2 * h + b),  act = relu if RELU != 0.
__global__ void k_finalize(float* __restrict__ agg,
                           const float* __restrict__ h,
                           const float* __restrict__ dinv,
                           const float* __restrict__ b,
                           int relu) {
    int stride = gridDim.x * blockDim.x;
    for (int i = blockIdx.x * blockDim.x + threadIdx.x; i < N_NODES * HID; i += stride) {
        const int n = i >> 6;      // /HID
        const int f = i & 63;      // %HID
        const float di = dinv[n];
        float v = agg[i] + di * di * h[i] + b[f];
        agg[i] = relu ? fmaxf(v, 0.0f) : v;
    }
}

// ---------------------------------------------------------------------------
// FC: out[n][o] = sum_k h[n][k] * Wfc[k][o] + bfc[o], OUT_DIM = 5.
__global__ void k_fc(const float* __restrict__ h,
                     const float* __restrict__ Wfc,
                     const float* __restrict__ bfc,
                     float* __restrict__ out) {
    int n = blockIdx.x * blockDim.x + threadIdx.x;
    if (n >= N_NODES) return;
    float acc[OUT_DIM];
    #pragma unroll
    for (int o = 0; o < OUT_DIM; ++o) acc[o] = bfc[o];
    const float* hr = h + (size_t)n * HID;
    #pragma unroll 4
    for (int k = 0; k < HID; ++k) {
        const float xv = hr[k];
        #pragma unroll
        for (int o = 0; o < OUT_DIM; ++o)
            acc[o] = fmaf(xv, Wfc[k * OUT_DIM + o], acc[o]);
    }
    float* orow = out + (size_t)n * OUT_DIM;
    #pragma unroll
    for (int o = 0; o < OUT_DIM; ++o) orow[o] = acc[o];
}

// ---------------------------------------------------------------------------
extern "C" void kernel_launch(void* const* d_in, const int* in_sizes, int n_in,
                              void* d_out, int out_size, void* d_ws, size_t ws_size,
                              hipStream_t stream) {
    const float* x   = (const float*)d_in[0];
    const int*   ei  = (const int*)  d_in[1];   // [2, E]
    const float* ew  = (const float*)d_in[2];
    const float* W1  = (const float*)d_in[3];
    const float* b1  = (const float*)d_in[4];
    const float* W2  = (const float*)d_in[5];
    const float* b2  = (const float*)d_in[6];
    const float* Wfc = (const float*)d_in[7];
    const float* bfc = (const float*)d_in[8];
    float* out = (float*)d_out;

    const int* row = ei;            // source nodes
    const int* col = ei + N_EDGES;  // target nodes

    // Workspace layout: dinv | bufH | bufA
    char* ws = (char*)d_ws;
    size_t off = 0;
    float* dinv = (float*)(ws + off); off += ((size_t)N_NODES * 4 + 255) & ~(size_t)255;
    float* bufH = (float*)(ws + off); off += (size_t)N_NODES * HID * 4;
    float* bufA = (float*)(ws + off); off += (size_t)N_NODES * HID * 4;
    (void)ws_size; (void)n_in; (void)in_sizes; (void)out_size;

    const int TPB = 256;
    const int nodeBlocks = (N_NODES + TPB - 1) / TPB;               // 391
    const int featElems  = N_NODES * HID;                           // 6.4M
    const int gemmBlocks = ((N_NODES / 16) + (TPB / 32) - 1) / (TPB / 32); // 782

    // 1) degree (with self-loop weight 1)
    k_init_deg<<<nodeBlocks, TPB, 0, stream>>>(dinv);
    k_deg_scatter<<<1024, TPB, 0, stream>>>(col, ew, dinv);
    k_dinv<<<nodeBlocks, TPB, 0, stream>>>(dinv);

    // 2) layer 1: h = x @ W1 ; aggregate ; relu(+ self-loop + b1)
    k_gemm_n64<<<gemmBlocks, TPB, 0, stream>>>(x, W1, bufH, N_NODES, IN_DIM);
    k_zero<<<4096, TPB, 0, stream>>>(bufA, featElems);
    k_edge_scatter<<<2048, TPB, 0, stream>>>(row, col, ew, dinv, bufH, bufA);
    k_finalize<<<4096, TPB, 0, stream>>>(bufA, bufH, dinv, b1, 1);

    // 3) layer 2: h = agg @ W2 ; aggregate ; relu(+ self-loop + b2)
    k_gemm_n64<<<gemmBlocks, TPB, 0, stream>>>(bufA, W2, bufH, N_NODES, HID);
    k_zero<<<4096, TPB, 0, stream>>>(bufA, featElems);
    k_edge_scatter<<<2048, TPB, 0, stream>>>(row, col, ew, dinv, bufH, bufA);
    k_finalize<<<4096, TPB, 0, stream>>>(bufA, bufH, dinv, b2, 1);

    // 4) FC head
    k_fc<<<nodeBlocks, TPB, 0, stream>>>(bufA, Wfc, bfc, out);
}